// DualThresholdSelfregulatingIntegrate_82729660056218
// MI455X (gfx1250) — compile-verified
//
#include <hip/hip_runtime.h>
#include <math.h>

// Reference: spikes = (floor(v0 + cumsum(relu(x)*dt)) - floor(prev)) / dt
//   inputs        [16, 2048, 1024] f32
//   initial_state [16, 1024]       f32
//   output        [16, 2048, 1024] f32
// Memory-bound scan: 268 MB @ 23.3 TB/s ~= 11.5 us floor. One workgroup owns
// (b, 32-channel tile, full T=2048); the input slab is staged HBM->LDS exactly
// once with gfx1250 async-to-LDS (ASYNCcnt), then a hierarchical chunk scan
// runs entirely out of LDS and streams non-temporal stores back.

#define DT_F 0.001f

namespace cfg {
constexpr int B = 16;
constexpr int T = 2048;
constexpr int D = 1024;
constexpr int CH  = 32;              // channels per workgroup
constexpr int NCK = 16;              // time chunks per workgroup
constexpr int CKL = T / NCK;         // 128 timesteps per chunk
constexpr int THREADS = CH * NCK;    // 512 threads = 16 wave32
constexpr int GPT = CH / 4;          // float4 groups per time row (== 8)
constexpr int NCOPY = T * GPT;       // b128 copies per block (16384)
constexpr size_t SMEM_BYTES = (size_t)(T * CH + NCK * CH) * sizeof(float); // 258 KB < 320 KB
static_assert(GPT == 8, "copy index math assumes 8 float4 groups per row");
static_assert(NCOPY % THREADS == 0, "uniform trip count");
}

// ---- CDNA5 async global->LDS path (ASYNCcnt), with safe fallback ----------
#if __has_builtin(__builtin_amdgcn_global_load_async_to_lds_b128)
#define USE_ASYNC_LDS 1
#else
#define USE_ASYNC_LDS 0
#endif

#if USE_ASYNC_LDS
typedef int v4i __attribute__((ext_vector_type(4)));
typedef __attribute__((address_space(1))) v4i* gv4i_ptr;   // global
typedef __attribute__((address_space(3))) v4i* lv4i_ptr;   // LDS

__device__ __forceinline__ void async_copy_b128(const float* gsrc, float* ldst) {
  // global -> LDS, 16 bytes per lane, tracked by ASYNCcnt
  __builtin_amdgcn_global_load_async_to_lds_b128(
      (gv4i_ptr)(v4i*)(void*)const_cast<float*>(gsrc),
      (lv4i_ptr)(v4i*)(void*)ldst,
      /*imm offset*/0, /*cpol*/0);
}
#endif

__device__ __forceinline__ void wait_asynccnt0() {
#if __has_builtin(__builtin_amdgcn_s_wait_asynccnt)
  __builtin_amdgcn_s_wait_asynccnt(0);
#else
  asm volatile("s_wait_asynccnt 0x0" ::: "memory");
#endif
}

__global__ __launch_bounds__(cfg::THREADS)
void spike_scan_kernel(const float* __restrict__ gin_all,
                       const float* __restrict__ v0,
                       float* __restrict__ gout_all) {
  using namespace cfg;
  extern __shared__ float smem[];
  float* s_x   = smem;            // [T][CH]  raw inputs (bank-conflict-free reads)
  float* s_sum = smem + T * CH;   // [NCK][CH] per-chunk rate sums

  const int b   = blockIdx.y;
  const int d0  = blockIdx.x * CH;
  const int tid = threadIdx.x;

  const float* gin  = gin_all  + ((size_t)b * T) * D + d0;
  float*       gout = gout_all + ((size_t)b * T) * D + d0;

  // ---- Phase 0: stage the [T x CH] slab into LDS (HBM read happens ONCE) --
  // LDS is a contiguous image of the slab: byte offset of copy f is f*16.
#if USE_ASYNC_LDS
  #pragma unroll 4
  for (int f = tid; f < NCOPY; f += THREADS) {
    const int t = f >> 3;            // time row
    const int g = f & (GPT - 1);     // float4 group within row
    async_copy_b128(gin + (size_t)t * D + g * 4, s_x + f * 4);
  }
  wait_asynccnt0();          // my wave's async copies are in LDS
  __syncthreads();           // everyone's copies visible
#else
  #pragma unroll 4
  for (int f = tid; f < NCOPY; f += THREADS) {
    const int t = f >> 3;
    const int g = f & (GPT - 1);
    const float4 v = *(const float4*)(gin + (size_t)t * D + g * 4);
    *(float4*)(s_x + f * 4) = v;     // ds_store_b128
  }
  __syncthreads();
#endif

  // ---- Phase 1: per-chunk rate sums -------------------------------------
  const int d  = tid & (CH - 1);   // channel within tile
  const int ck = tid / CH;         // time chunk
  {
    float sum = 0.0f;
    const float* p = s_x + (size_t)ck * CKL * CH + d;
    #pragma unroll 8
    for (int i = 0; i < CKL; ++i)
      sum += fmaxf(p[i * CH], 0.0f);     // ds_load_b32, 32 distinct banks
    s_sum[ck * CH + d] = sum * DT_F;
  }
  __syncthreads();

  // ---- Phase 2: exclusive chunk prefix + emit spikes --------------------
  float off = v0[(size_t)b * D + d0 + d];
  for (int j = 0; j < ck; ++j)
    off += s_sum[j * CH + d];

  // off == v0 + cumsum(rates) through end of previous chunk, so floor(off)
  // is exactly the reference's floor(c_{t-1}) at the chunk boundary
  // (and floor(v0) for chunk 0).
  constexpr float INV_DT = 1.0f / DT_F;   // folded at compile time
  float prev = floorf(off);
  float c = off;
  const float* p = s_x + (size_t)ck * CKL * CH + d;
  float*       q = gout + (size_t)ck * CKL * D + d;
  #pragma unroll 8
  for (int i = 0; i < CKL; ++i) {
    c += fmaxf(p[i * CH], 0.0f) * DT_F;   // v_max + v_fmac
    const float fl = floorf(c);
    // write-once stream: non-temporal, contiguous 128B per wave store
    __builtin_nontemporal_store((fl - prev) * INV_DT, q + (size_t)i * D);
    prev = fl;
  }
}

extern "C" void kernel_launch(void* const* d_in, const int* in_sizes, int n_in,
                              void* d_out, int out_size, void* d_ws, size_t ws_size,
                              hipStream_t stream) {
  (void)in_sizes; (void)n_in; (void)out_size; (void)d_ws; (void)ws_size;
  const float* inputs = (const float*)d_in[0];   // [16, 2048, 1024]
  const float* init   = (const float*)d_in[1];   // [16, 1024]
  float*       out    = (float*)d_out;           // [16, 2048, 1024]

  dim3 grid(cfg::D / cfg::CH, cfg::B);           // 32 x 16 = 512 blocks
  spike_scan_kernel<<<grid, dim3(cfg::THREADS), cfg::SMEM_BYTES, stream>>>(
      inputs, init, out);
}